// Model_8572754723151
// MI455X (gfx1250) — compile-verified
//
#include <hip/hip_runtime.h>
#include <hip/hip_bf16.h>
#include <math.h>

#define D 64

typedef float v2f __attribute__((ext_vector_type(2)));
typedef float v8f __attribute__((ext_vector_type(8)));

// Guaranteed-hardware f32 atomics (no-return form, STOREcnt-tracked).
// CDNA5: GLOBAL_ATOMIC_ADD_F32 (op 86), GLOBAL_ATOMIC_MAX_NUM_F32 (op 82).
__device__ __forceinline__ void atom_add_f32(float* p, float v) {
  asm volatile("global_atomic_add_f32 %0, %1, off scope:SCOPE_DEV"
               :: "v"(p), "v"(v) : "memory");
}
__device__ __forceinline__ void atom_max_f32(float* p, float v) {
  asm volatile("global_atomic_max_num_f32 %0, %1, off scope:SCOPE_DEV"
               :: "v"(p), "v"(v) : "memory");
}

// ---------------- gather: h0 = embed[node_ids]; also write out[:,0:64] ------
__global__ __launch_bounds__(256) void gather_kernel(
    const float* __restrict__ table, const int* __restrict__ ids,
    float* __restrict__ h0, float* __restrict__ out, int n) {
  int row = blockIdx.x * 8 + (threadIdx.x >> 5);
  int lane = threadIdx.x & 31;
  if (row >= n) return;
  int nid = ids[row];
  float2 v = *(const float2*)(table + (size_t)nid * D + lane * 2);
  *(float2*)(h0 + (size_t)row * D + lane * 2) = v;
  *(float2*)(out + (size_t)row * (3 * D) + lane * 2) = v;
}

// ---------------- init m=-inf, s=0, hout=0 ---------------------------------
__global__ __launch_bounds__(256) void init_kernel(
    float* __restrict__ m, float* __restrict__ s,
    float* __restrict__ hout, int total) {
  int t = blockIdx.x * blockDim.x + threadIdx.x;
  if (t >= total) return;
  m[t] = -INFINITY;
  s[t] = 0.0f;
  hout[t] = 0.0f;
}

// ---------------- feat = relu(H @ W^T + b) via V_WMMA_F32_16X16X4_F32 ------
// One wave computes a 16x64 output tile. A = H tile (16x4 per step),
// B = W^T tile (4x16 per step): B[k][j] = W[j][k].
__global__ __launch_bounds__(256) void gemm_relu_wmma(
    const float* __restrict__ H, const float* __restrict__ W,
    const float* __restrict__ bias, float* __restrict__ outF, int n) {
  int wave = blockIdx.x * 8 + (threadIdx.x >> 5);
  int lane = threadIdx.x & 31;
  int row0 = wave * 16;
  if (row0 >= n) return;
  int lmod = lane & 15;
  bool hi = lane >= 16;
  int mrow = row0 + lmod;
  bool fullTile = (row0 + 16 <= n);
  // 16x4 f32 A layout: lanes 0-15 hold K=0,1 ; lanes 16-31 hold K=2,3
  const float* arow = H + (size_t)(mrow < n ? mrow : n - 1) * D + (hi ? 2 : 0);
#pragma unroll
  for (int jt = 0; jt < 4; ++jt) {
    const float* brow = W + (size_t)(jt * 16 + lmod) * D + (hi ? 2 : 0);
    float bj = bias[jt * 16 + lmod];
    v8f acc;
#pragma unroll
    for (int v = 0; v < 8; ++v) acc[v] = bj;  // C/D: same N per lane -> bias bcast
#pragma unroll
    for (int k0 = 0; k0 < 16; ++k0) {
      float2 af = *(const float2*)(arow + k0 * 4);
      float2 bf = *(const float2*)(brow + k0 * 4);
      v2f av; av[0] = af.x; av[1] = af.y;
      v2f bv; bv[0] = bf.x; bv[1] = bf.y;
      acc = __builtin_amdgcn_wmma_f32_16x16x4_f32(false, av, false, bv,
                                                  (short)0, acc, false, false);
    }
    // D layout: VGPR v -> row (row0+v) for lanes<16, row (row0+v+8) for lanes>=16
    float* obase = outF + (size_t)(row0 + (hi ? 8 : 0)) * D + jt * 16 + lmod;
    if (fullTile) {
#pragma unroll
      for (int v = 0; v < 8; ++v)
        obase[(size_t)v * D] = fmaxf(acc[v], 0.0f);
    } else {
#pragma unroll
      for (int v = 0; v < 8; ++v) {
        int orow = row0 + v + (hi ? 8 : 0);
        if (orow < n)
          outF[(size_t)orow * D + jt * 16 + lmod] = fmaxf(acc[v], 0.0f);
      }
    }
  }
}

// ---------------- pass 1: m[dst] = segment_max(h[src]*h[dst]) --------------
__global__ __launch_bounds__(256) void edge_max_kernel(
    const float* __restrict__ h, const int* __restrict__ src,
    const int* __restrict__ dst, float* __restrict__ m, int E) {
  int e = blockIdx.x * 8 + (threadIdx.x >> 5);
  int lane = threadIdx.x & 31;
  if (e >= E) return;
  int s = src[e], d = dst[e];
  int c = lane * 2;
  float2 hs = *(const float2*)(h + (size_t)s * D + c);
  float2 hd = *(const float2*)(h + (size_t)d * D + c);
  float* mp = m + (size_t)d * D + c;
  atom_max_f32(mp, hs.x * hd.x);
  atom_max_f32(mp + 1, hs.y * hd.y);
}

// ---------------- pass 2: s[dst] += exp(e - m[dst]) ------------------------
__global__ __launch_bounds__(256) void edge_sum_kernel(
    const float* __restrict__ h, const int* __restrict__ src,
    const int* __restrict__ dst, const float* __restrict__ m,
    float* __restrict__ sbuf, int E) {
  int e = blockIdx.x * 8 + (threadIdx.x >> 5);
  int lane = threadIdx.x & 31;
  if (e >= E) return;
  int s = src[e], d = dst[e];
  int c = lane * 2;
  float2 hs = *(const float2*)(h + (size_t)s * D + c);
  float2 hd = *(const float2*)(h + (size_t)d * D + c);
  float2 mm = *(const float2*)(m + (size_t)d * D + c);
  float* sp = sbuf + (size_t)d * D + c;
  atom_add_f32(sp, __expf(hs.x * hd.x - mm.x));
  atom_add_f32(sp + 1, __expf(hs.y * hd.y - mm.y));
}

// ---------------- pass 3: hout[dst] += feat[src] * alpha -------------------
__global__ __launch_bounds__(256) void edge_msg_kernel(
    const float* __restrict__ h, const int* __restrict__ src,
    const int* __restrict__ dst, const float* __restrict__ m,
    const float* __restrict__ sbuf, const float* __restrict__ feat,
    float* __restrict__ hout, int E) {
  int e = blockIdx.x * 8 + (threadIdx.x >> 5);
  int lane = threadIdx.x & 31;
  if (e >= E) return;
  int s = src[e], d = dst[e];
  int c = lane * 2;
  float2 hs = *(const float2*)(h + (size_t)s * D + c);
  float2 hd = *(const float2*)(h + (size_t)d * D + c);
  float2 mm = *(const float2*)(m + (size_t)d * D + c);
  float2 ss = *(const float2*)(sbuf + (size_t)d * D + c);
  float2 fs = *(const float2*)(feat + (size_t)s * D + c);
  float a0 = __expf(hs.x * hd.x - mm.x) / ss.x;
  float a1 = __expf(hs.y * hd.y - mm.y) / ss.y;
  float* op = hout + (size_t)d * D + c;
  atom_add_f32(op, fs.x * a0);
  atom_add_f32(op + 1, fs.y * a1);
}

// ---------------- l2-normalize row, write to out column block --------------
__global__ __launch_bounds__(256) void norm_out_kernel(
    const float* __restrict__ h, float* __restrict__ out, int col_off, int n) {
  int row = blockIdx.x * 8 + (threadIdx.x >> 5);
  int lane = threadIdx.x & 31;
  if (row >= n) return;
  float2 x = *(const float2*)(h + (size_t)row * D + lane * 2);
  float ss = x.x * x.x + x.y * x.y;
#pragma unroll
  for (int msk = 16; msk >= 1; msk >>= 1) ss += __shfl_xor(ss, msk, 32);
  float inv = 1.0f / fmaxf(sqrtf(ss), 1e-12f);
  float2 r; r.x = x.x * inv; r.y = x.y * inv;
  *(float2*)(out + (size_t)row * (3 * D) + col_off + lane * 2) = r;
}

extern "C" void kernel_launch(void* const* d_in, const int* in_sizes, int n_in,
                              void* d_out, int out_size, void* d_ws, size_t ws_size,
                              hipStream_t stream) {
  const int* node_ids = (const int*)d_in[0];
  const int* src      = (const int*)d_in[1];
  const int* dst      = (const int*)d_in[2];
  const float* table  = (const float*)d_in[3];
  const float* Ws0    = (const float*)d_in[4];
  const float* bs0    = (const float*)d_in[5];
  // d_in[6], d_in[7] = W_dst_0/b_dst_0: dead in the reference output
  const float* Ws1    = (const float*)d_in[8];
  const float* bs1    = (const float*)d_in[9];
  float* out = (float*)d_out;

  int N = in_sizes[0];
  int E = in_sizes[1];
  size_t ND = (size_t)N * D;

  float* h0   = (float*)d_ws;
  float* h1   = h0 + ND;
  float* h2   = h1 + ND;
  float* feat = h2 + ND;
  float* mbuf = feat + ND;
  float* sbuf = mbuf + ND;

  dim3 blk(256);
  int nodeBlocks = (N + 7) / 8;             // 8 waves/block, 1 wave per row
  int edgeBlocks = (E + 7) / 8;             // 1 wave per edge
  int initBlocks = (int)((ND + 255) / 256);
  int gemmBlocks = ((N + 15) / 16 + 7) / 8; // 1 wave per 16-row tile

  // h0 = embed[node_ids]; out[:,0:64] = h0
  gather_kernel<<<nodeBlocks, blk, 0, stream>>>(table, node_ids, h0, out, N);

  // ---- layer 1: h0 -> h1 ----
  init_kernel<<<initBlocks, blk, 0, stream>>>(mbuf, sbuf, h1, (int)ND);
  gemm_relu_wmma<<<gemmBlocks, blk, 0, stream>>>(h0, Ws0, bs0, feat, N);
  edge_max_kernel<<<edgeBlocks, blk, 0, stream>>>(h0, src, dst, mbuf, E);
  edge_sum_kernel<<<edgeBlocks, blk, 0, stream>>>(h0, src, dst, mbuf, sbuf, E);
  edge_msg_kernel<<<edgeBlocks, blk, 0, stream>>>(h0, src, dst, mbuf, sbuf, feat, h1, E);
  norm_out_kernel<<<nodeBlocks, blk, 0, stream>>>(h1, out, D, N);

  // ---- layer 2: h1 (unnormalized) -> h2 ----
  init_kernel<<<initBlocks, blk, 0, stream>>>(mbuf, sbuf, h2, (int)ND);
  gemm_relu_wmma<<<gemmBlocks, blk, 0, stream>>>(h1, Ws1, bs1, feat, N);
  edge_max_kernel<<<edgeBlocks, blk, 0, stream>>>(h1, src, dst, mbuf, E);
  edge_sum_kernel<<<edgeBlocks, blk, 0, stream>>>(h1, src, dst, mbuf, sbuf, E);
  edge_msg_kernel<<<edgeBlocks, blk, 0, stream>>>(h1, src, dst, mbuf, sbuf, feat, h2, E);
  norm_out_kernel<<<nodeBlocks, blk, 0, stream>>>(h2, out, 2 * D, N);
}